// HiddenStateMLPPooling_46634754900233
// MI455X (gfx1250) — compile-verified
//
#include <hip/hip_runtime.h>

typedef __attribute__((ext_vector_type(2))) float v2f;
typedef __attribute__((ext_vector_type(8))) float v8f;

#define N_AGENTS 1024
#define HID_IN   128   // HIDDEN_DIM
#define HID_OUT  64    // MLP_HIDDEN
#define POOL_DIM 128   // MLP_DIM
#define ODIM     128   // OUT_DIM

// Monotone float <-> uint mapping so unsigned atomicMax == float max.
__device__ __forceinline__ unsigned enc_f32(float f) {
  unsigned b = __float_as_uint(f);
  return (b & 0x80000000u) ? ~b : (b | 0x80000000u);
}
__device__ __forceinline__ float dec_f32(unsigned k) {
  return (k & 0x80000000u) ? __uint_as_float(k & 0x7FFFFFFFu)
                           : __uint_as_float(~k);
}

// ---------------------------------------------------------------------------
// Kernel A: hidden = relu(hs @ W_hid + b_hid)  [1024x64]; column-max -> keys.
// One 16x16 output tile per wave, K=128 via 32x v_wmma_f32_16x16x4_f32.
// ---------------------------------------------------------------------------
__global__ __launch_bounds__(256) void khidden(const float* __restrict__ hs,
                                               const float* __restrict__ W_hid,
                                               const float* __restrict__ b_hid,
                                               unsigned* __restrict__ hidKeys) {
  const int lane  = threadIdx.x & 31;
  const int wave  = threadIdx.x >> 5;
  const int tile  = blockIdx.x * 8 + wave;   // 256 tiles = 64 Mtiles x 4 Ntiles
  const int mtile = tile >> 2;
  const int ntile = tile & 3;
  const int lrow  = lane & 15;
  const int kh    = (lane >> 4) << 1;        // K sub-offset: 0 or 2

  const float* arow = hs + (mtile * 16 + lrow) * HID_IN;
  const int    ncol = ntile * 16 + lrow;

  v8f acc = {};
#pragma unroll
  for (int k = 0; k < HID_IN; k += 4) {
    v2f a, b;
    a[0] = arow[k + kh + 0];                       // A[m, k+kh+v]
    a[1] = arow[k + kh + 1];
    b[0] = W_hid[(k + kh + 0) * HID_OUT + ncol];   // B[k+kh+v, n]
    b[1] = W_hid[(k + kh + 1) * HID_OUT + ncol];
    acc = __builtin_amdgcn_wmma_f32_16x16x4_f32(false, a, false, b,
                                                (short)0, acc, false, false);
  }

  // bias + relu, then max over the 16 rows of this tile's column ncol.
  const float bias = b_hid[ncol];
  float cmax = 0.0f;                          // relu floor
#pragma unroll
  for (int v = 0; v < 8; ++v) cmax = fmaxf(cmax, acc[v] + bias);
  cmax = fmaxf(cmax, __shfl_xor(cmax, 16));   // merge rows M and M+8 halves
  if (lane < 16) atomicMax(&hidKeys[ncol], enc_f32(cmax));
}

// ---------------------------------------------------------------------------
// Kernel B: per-agent p/q projections, cluster-wide max via LDS, build pooled.
// sp_max[i,d] = relu(Pmax[d] - p[i,d] + b_sp[d])  (exact O(N) rewrite).
// ---------------------------------------------------------------------------
__global__ __launch_bounds__(1024) void kpool(const float* __restrict__ obs1,
                                              const float* __restrict__ obs,
                                              const float* __restrict__ W_sp,
                                              const float* __restrict__ b_sp,
                                              const float* __restrict__ W_vel,
                                              const float* __restrict__ b_vel,
                                              const unsigned* __restrict__ hidKeys,
                                              float* __restrict__ pooled) {
  __shared__ unsigned smax[64];               // [0..31]=Pmax keys, [32..63]=Qmax
  const int i = threadIdx.x;                  // agent row, 0..1023
  if (i < 64) smax[i] = 0u;
  __syncthreads();

  const float o0 = obs[i * 2 + 0], o1 = obs[i * 2 + 1];
  const float v0 = o0 - obs1[i * 2 + 0], v1 = o1 - obs1[i * 2 + 1];

  float p[32], q[32];
#pragma unroll
  for (int d = 0; d < 32; ++d) {
    p[d] = o0 * W_sp[d] + o1 * W_sp[32 + d];
    q[d] = 4.0f * (v0 * W_vel[d] + v1 * W_vel[32 + d]);
    float pm = p[d], qm = q[d];
#pragma unroll
    for (int off = 16; off > 0; off >>= 1) {  // wave32 max-reduce
      pm = fmaxf(pm, __shfl_xor(pm, off));
      qm = fmaxf(qm, __shfl_xor(qm, off));
    }
    if ((i & 31) == 0) {
      atomicMax(&smax[d], enc_f32(pm));
      atomicMax(&smax[32 + d], enc_f32(qm));
    }
  }
  __syncthreads();

  float* row = pooled + i * POOL_DIM;
#pragma unroll
  for (int d = 0; d < 32; ++d) {
    row[d]      = fmaxf(dec_f32(smax[d])      - p[d] + b_sp[d],  0.0f);
    row[32 + d] = fmaxf(dec_f32(smax[32 + d]) - q[d] + b_vel[d], 0.0f);
  }
#pragma unroll
  for (int d = 0; d < 64; ++d) row[64 + d] = dec_f32(hidKeys[d]);
}

// ---------------------------------------------------------------------------
// Kernel C: out = pooled @ W_out + b_out  [1024x128]@[128x128], fp32 WMMA.
// ---------------------------------------------------------------------------
__global__ __launch_bounds__(256) void kout(const float* __restrict__ pooled,
                                            const float* __restrict__ W_out,
                                            const float* __restrict__ b_out,
                                            float* __restrict__ out) {
  const int lane  = threadIdx.x & 31;
  const int wave  = threadIdx.x >> 5;
  const int tile  = blockIdx.x * 8 + wave;   // 512 tiles = 64 Mtiles x 8 Ntiles
  const int mtile = tile >> 3;
  const int ntile = tile & 7;
  const int lrow  = lane & 15;
  const int kh    = (lane >> 4) << 1;

  const float* arow = pooled + (mtile * 16 + lrow) * POOL_DIM;
  const int    ncol = ntile * 16 + lrow;

  v8f acc = {};
#pragma unroll
  for (int k = 0; k < POOL_DIM; k += 4) {
    v2f a, b;
    a[0] = arow[k + kh + 0];
    a[1] = arow[k + kh + 1];
    b[0] = W_out[(k + kh + 0) * ODIM + ncol];
    b[1] = W_out[(k + kh + 1) * ODIM + ncol];
    acc = __builtin_amdgcn_wmma_f32_16x16x4_f32(false, a, false, b,
                                                (short)0, acc, false, false);
  }

  const float bias  = b_out[ncol];
  const int   mrow0 = mtile * 16 + ((lane >> 4) << 3);  // C/D: M = v + 8*(lane/16)
#pragma unroll
  for (int v = 0; v < 8; ++v)
    out[(mrow0 + v) * ODIM + ntile * 16 + lrow] = acc[v] + bias;
}

// ---------------------------------------------------------------------------
extern "C" void kernel_launch(void* const* d_in, const int* in_sizes, int n_in,
                              void* d_out, int out_size, void* d_ws, size_t ws_size,
                              hipStream_t stream) {
  const float* hs    = (const float*)d_in[0];
  const float* obs1  = (const float*)d_in[1];
  const float* obs   = (const float*)d_in[2];
  const float* W_sp  = (const float*)d_in[3];
  const float* b_sp  = (const float*)d_in[4];
  const float* W_vel = (const float*)d_in[5];
  const float* b_vel = (const float*)d_in[6];
  const float* W_hid = (const float*)d_in[7];
  const float* b_hid = (const float*)d_in[8];
  const float* W_out = (const float*)d_in[9];
  const float* b_out = (const float*)d_in[10];

  float*    pooled  = (float*)d_ws;                              // 1024*128 f32
  unsigned* hidKeys = (unsigned*)(pooled + N_AGENTS * POOL_DIM); // 64 keys

  hipMemsetAsync(hidKeys, 0, HID_OUT * sizeof(unsigned), stream);
  khidden<<<32, 256, 0, stream>>>(hs, W_hid, b_hid, hidKeys);
  kpool<<<1, 1024, 0, stream>>>(obs1, obs, W_sp, b_sp, W_vel, b_vel, hidKeys, pooled);
  kout<<<64, 256, 0, stream>>>(pooled, W_out, b_out, (float*)d_out);
}